// QConv2d_24111946399868
// MI455X (gfx1250) — compile-verified
//
#include <hip/hip_runtime.h>
#include <hip/hip_bf16.h>

// Problem constants (from reference): B=256, D=8, C_IN=2, C_OUT=4
// C = C_OUT*D*D = 256, Ksub = C_IN*D*D = 128.
#define NBATCH 256
#define CDIM   256
#define KSUB   128

// Padded LDS row strides (floats), chosen for bank-conflict-free access:
//  RHO_S = 144: 2*144 mod 64 = 32 -> the two half-wave K-rows of a B-fragment
//              load hit disjoint 16-bank ranges.
//  TT_S  = 137: odd -> Phase-A tile stores across lanes 0..15 hit distinct
//              banks; 2*137 mod 64 = 18 -> half-wave rows of Phase-B A-loads
//              are bank-disjoint.
#define RHO_S  144
#define TT_S   137

typedef __attribute__((ext_vector_type(2))) float v2f;
typedef __attribute__((ext_vector_type(8))) float v8f;

// -------------------------------------------------------------------------
// Setup: UT[k][p] = U_sub[p][k] = kron(uc,kron(ux,uy))[p, 128+k]
// (serves as the A operand of GEMM1 and the B operand of GEMM2, both with
//  lane-contiguous loads).
// -------------------------------------------------------------------------
__global__ void build_ut_kernel(const float* __restrict__ ux,
                                const float* __restrict__ uy,
                                const float* __restrict__ uc,
                                float* __restrict__ UT) { // [128*256]
    int idx = blockIdx.x * blockDim.x + threadIdx.x;      // 0 .. 32767
    if (idx >= CDIM * KSUB) return;
    int p = idx / KSUB;           // output row index 0..255
    int q = idx % KSUB;           // sub-column 0..127 (global col = 128+q)
    int c1 = p >> 6;
    int i1 = (p >> 3) & 7;
    int j1 = p & 7;
    int c2 = 2 + (q >> 6);        // only channels 2,3 are nonzero
    int i2 = (q >> 3) & 7;
    int j2 = q & 7;
    UT[q * CDIM + p] = uc[c1 * 4 + c2] * ux[i1 * 8 + i2] * uy[j1 * 8 + j2];
}

// -------------------------------------------------------------------------
// Main: 2 workgroups per batch (half = output rows [half*128, half*128+128)).
//   LDS: sRho (128x128, rows padded to 144) = rho[b]
//        sTt  (128x128, rows padded to 137) = (U_half @ rho)^T, i.e. sTt[k][p]
//   out[b][half rows] = T_half @ UT, via V_WMMA_F32_16X16X4_F32.
// -------------------------------------------------------------------------
__global__ void __launch_bounds__(256, 2)
qconv_wmma_kernel(const float* __restrict__ rho,   // [256][128][128]
                  const float* __restrict__ UT,    // [128][256]
                  float* __restrict__ out) {       // [256][256][256]
    extern __shared__ float smem[];
    float* sRho = smem;                 // 128 * RHO_S floats
    float* sTt  = smem + KSUB * RHO_S;  // 128 * TT_S floats

    const int b    = blockIdx.x >> 1;
    const int half = blockIdx.x & 1;
    const int tid  = threadIdx.x;
    const int lane = tid & 31;
    const int w    = tid >> 5;          // wave id 0..7
    const int l16  = lane & 15;
    const int hl   = lane >> 4;         // half-wave select

    // ---- load rho[b] into padded LDS rows, coalesced float4 ----
    {
        const float4* src = (const float4*)(rho + (size_t)b * KSUB * KSUB);
        float4* dst = (float4*)sRho;    // padded row = 36 float4s of data + pad
        #pragma unroll
        for (int i = 0; i < (KSUB * KSUB / 4) / 256; ++i) {
            int g    = tid + i * 256;   // float4 index, 0..4095
            int row  = g >> 5;          // /32 float4s per logical row
            int col4 = g & 31;
            dst[row * (RHO_S / 4) + col4] = src[g];
        }
    }
    __syncthreads();

    // ====== Phase A: T_half(128x128) = U_sub[half rows] @ sRho(128x128) ====
    // 8x8 tiles of 16x16; wave w handles tm = w, tn = 0..7.
    {
        const int prow = half * KSUB + w * 16 + l16;   // global U row (A M-index)
        v8f acc[8] = {v8f{}, v8f{}, v8f{}, v8f{}, v8f{}, v8f{}, v8f{}, v8f{}};
        for (int kk = 0; kk < KSUB; kk += 4) {
            // A fragment: A[p][k] = UT[k][p], lane-contiguous across l16
            v2f a;
            a.x = UT[(kk + 2 * hl)     * CDIM + prow];
            a.y = UT[(kk + 2 * hl + 1) * CDIM + prow];
            #pragma unroll
            for (int tn = 0; tn < 8; ++tn) {
                v2f bf;   // B fragment: rho rows (kk+2hl, +1), cols tn*16+l16
                bf.x = sRho[(kk + 2 * hl)     * RHO_S + tn * 16 + l16];
                bf.y = sRho[(kk + 2 * hl + 1) * RHO_S + tn * 16 + l16];
                acc[tn] = __builtin_amdgcn_wmma_f32_16x16x4_f32(
                    false, a, false, bf, (short)0, acc[tn], false, false);
            }
        }
        // Store transposed: sTt[r][p_local], r = tn*16+l16, p_local = w*16+M
        #pragma unroll
        for (int tn = 0; tn < 8; ++tn) {
            #pragma unroll
            for (int v = 0; v < 8; ++v)
                sTt[(tn * 16 + l16) * TT_S + w * 16 + v + 8 * hl] = acc[tn][v];
        }
    }
    __syncthreads();

    // ====== Phase B: out[b][half rows](128x256) = T_half @ UT(128x256) =====
    // 8x16 tiles; wave w handles tm = w, tn = 0..15 in two halves of 8.
    float* outb = out + (size_t)b * CDIM * CDIM + (size_t)half * KSUB * CDIM;
    const int plocal = w * 16 + l16;                   // local T row (A M-index)
    for (int th = 0; th < 2; ++th) {
        v8f acc[8] = {v8f{}, v8f{}, v8f{}, v8f{}, v8f{}, v8f{}, v8f{}, v8f{}};
        for (int kk = 0; kk < KSUB; kk += 4) {
            // A fragment: T[p][k] = sTt[k][p], lane-contiguous across l16
            v2f a;
            a.x = sTt[(kk + 2 * hl)     * TT_S + plocal];
            a.y = sTt[(kk + 2 * hl + 1) * TT_S + plocal];
            #pragma unroll
            for (int tn8 = 0; tn8 < 8; ++tn8) {
                const int tn = th * 8 + tn8;
                v2f bf;   // B fragment: UT rows (kk+2hl, +1), lane-contiguous
                bf.x = UT[(kk + 2 * hl)     * CDIM + tn * 16 + l16];
                bf.y = UT[(kk + 2 * hl + 1) * CDIM + tn * 16 + l16];
                acc[tn8] = __builtin_amdgcn_wmma_f32_16x16x4_f32(
                    false, a, false, bf, (short)0, acc[tn8], false, false);
            }
        }
        #pragma unroll
        for (int tn8 = 0; tn8 < 8; ++tn8) {
            const int tn = th * 8 + tn8;
            #pragma unroll
            for (int v = 0; v < 8; ++v)
                outb[(w * 16 + v + 8 * hl) * CDIM + tn * 16 + l16] = acc[tn8][v];
        }
    }
}

// -------------------------------------------------------------------------
extern "C" void kernel_launch(void* const* d_in, const int* in_sizes, int n_in,
                              void* d_out, int out_size, void* d_ws, size_t ws_size,
                              hipStream_t stream) {
    (void)in_sizes; (void)n_in; (void)out_size; (void)ws_size;
    const float* rho = (const float*)d_in[0];   // [256*128*128]
    const float* ux  = (const float*)d_in[1];   // [8*8]
    const float* uy  = (const float*)d_in[2];   // [8*8]
    const float* uc  = (const float*)d_in[3];   // [4*4]
    float* out = (float*)d_out;                 // [256*256*256]

    float* UT = (float*)d_ws;                   // 128*256 floats = 128 KB

    build_ut_kernel<<<(CDIM * KSUB + 255) / 256, 256, 0, stream>>>(ux, uy, uc, UT);

    // ~140.5 KB dynamic LDS per workgroup -> 2 workgroups per 320 KB WGP
    // (4 waves/SIMD), above the 64 KB default cap so raise the attribute.
    const int lds_bytes = (KSUB * RHO_S + KSUB * TT_S) * (int)sizeof(float);
    (void)hipFuncSetAttribute((const void*)qconv_wmma_kernel,
                              hipFuncAttributeMaxDynamicSharedMemorySize, lds_bytes);

    qconv_wmma_kernel<<<NBATCH * 2, 256, lds_bytes, stream>>>(rho, UT, out);
}